// Index2Feat_module_4209067950243
// MI455X (gfx1250) — compile-verified
//
#include <hip/hip_runtime.h>
#include <hip/hip_bf16.h>

typedef __attribute__((ext_vector_type(2))) float v2f;
typedef __attribute__((ext_vector_type(8))) float v8f;

#define C_CH   256
#define H_SP   128
#define W_SP   128
#define N_KP   512
#define E_DIM  1024
#define KTOT   1024          // C*K*K
#define KC     512           // K-chunk per phase (2 phases)
#define LDS_STRIDE 516       // KC + 4 pad -> 516 % 64 == 4 -> conflict-free frag reads

// D(16e x 16s) = A(16e x 4k) * B(4k x 16s) + C, accumulated over k with
// v_wmma_f32_16x16x4_f32.  A rows come from W (row-major E x 1024), B columns
// are the gathered per-sample patch vectors staged in LDS.
__global__ __launch_bounds__(256, 2)
void index2feat_wmma(const float* __restrict__ img,
                     const float* __restrict__ Wm,
                     const float* __restrict__ bias,
                     const int*   __restrict__ bidx,
                     const int*   __restrict__ pxid,
                     const int*   __restrict__ pyid,
                     float* __restrict__ out)
{
    __shared__ float G[16 * LDS_STRIDE];   // 16 samples x 512 k (padded)

    const int t    = threadIdx.x;
    const int lane = t & 31;
    const int wv   = t >> 5;               // wave 0..7

    // ---- sample-tile decode: blockIdx.x in [0,1024) = (b*4+off)*32 + ntile
    const int st  = blockIdx.x;
    const int bo  = st >> 5;               // b*4 + off
    const int b   = bo >> 2;
    const int off = bo & 3;
    const int n0  = (st & 31) << 4;        // first keypoint of this 16-sample tile
    const int dy  = (off & 1) << 1;        // offsets in order (0,0),(2,0),(0,2),(2,2)
    const int dx  = off & 2;

    // ---- per-thread gather identity: thread t -> sample s = t&15
    const int s    = t & 15;
    const int pgrp = t >> 4;               // 0..15
    const int gi   = b * N_KP + n0 + s;
    const int bs   = bidx[gi];
    const int y    = 2 * pyid[gi] + dy;    // sampled conv-output row (0..128)
    const int x    = 2 * pxid[gi] + dx;    // sampled conv-output col (0..128)
    const float* imgb = img + (size_t)bs * (C_CH * H_SP * W_SP);

    // ---- per-wave WMMA setup: this wave owns filters [ebase, ebase+32)
    const int ebase = blockIdx.y * 256 + wv * 32;
    const int erow0 = ebase + (lane & 15);             // A frag: e row per lane
    const int khalf = (lane >> 4) << 1;                // k sub-offset 0 or 2
    const float* wrow0 = Wm + (size_t)erow0 * KTOT + khalf;
    const float* wrow1 = wrow0 + (size_t)16 * KTOT;    // second accumulator (+16 e)
    const float* gbp   = &G[(lane & 15) * LDS_STRIDE + khalf];  // B frag base

    v8f c0 = {};
    v8f c1 = {};

    for (int p = 0; p < 2; ++p) {
        // ---------- gather phase: fill G[s][0..511] for k in [p*512,(p+1)*512)
        // k = c*4 + ky*2 + kx ; element = img[bs, c, y+ky-1, x+kx-1] (0 if OOB)
        const int cbase = p << 7;          // p*128 channels per phase
        #pragma unroll 4
        for (int i = 0; i < 16; ++i) {
            const int pr = pgrp + (i << 4);        // 0..255 pair index
            const int cc = pr >> 1;                // channel within phase
            const int ky = pr & 1;
            const int yy = y + ky - 1;
            const float* rowp = imgb + ((size_t)(cbase + cc) * H_SP + yy) * W_SP;
            float v0 = 0.0f, v1 = 0.0f;
            if ((unsigned)yy < (unsigned)H_SP) {
                if ((unsigned)(x - 1) < (unsigned)W_SP) v0 = rowp[x - 1];
                if ((unsigned)x       < (unsigned)W_SP) v1 = rowp[x];
            }
            v2f pk; pk.x = v0; pk.y = v1;
            *(v2f*)&G[s * LDS_STRIDE + (cc << 2) + (ky << 1)] = pk;   // ds_store_b64
        }
        __syncthreads();

        // ---------- compute phase: 128 wmma k-steps over this 512-deep chunk
        const float* wp0 = wrow0 + (p << 9);
        const float* wp1 = wrow1 + (p << 9);
        #pragma unroll 8
        for (int k = 0; k < KC; k += 4) {
            v2f a0 = *(const v2f*)(wp0 + k);   // W[e0..e0+15][k..k+3] striped
            v2f a1 = *(const v2f*)(wp1 + k);
            v2f bf = *(const v2f*)(gbp + k);   // G[s][k..k+3] striped
            c0 = __builtin_amdgcn_wmma_f32_16x16x4_f32(
                     false, a0, false, bf, (short)0, c0, false, false);
            c1 = __builtin_amdgcn_wmma_f32_16x16x4_f32(
                     false, a1, false, bf, (short)0, c1, false, false);
        }
        __syncthreads();
    }

    // ---------- bias + store: D vgpr v -> e = ebase + v + 8*(lane>>4), n = lane&15
    const int nout = n0 + (lane & 15);
    const int mtop = (lane >> 4) << 3;
    const size_t obase = (size_t)bo * E_DIM * N_KP;    // (b*4+off)*1024*512
    #pragma unroll
    for (int v = 0; v < 8; ++v) {
        const int e0 = ebase + mtop + v;
        const int e1 = e0 + 16;
        out[obase + (size_t)e0 * N_KP + nout] = c0[v] + bias[e0];
        out[obase + (size_t)e1 * N_KP + nout] = c1[v] + bias[e1];
    }
}

extern "C" void kernel_launch(void* const* d_in, const int* in_sizes, int n_in,
                              void* d_out, int out_size, void* d_ws, size_t ws_size,
                              hipStream_t stream) {
    const float* img  = (const float*)d_in[0];   // (8,256,128,128) f32
    const float* Wm   = (const float*)d_in[1];   // (1024,256,2,2) f32 == (1024,1024)
    const float* bias = (const float*)d_in[2];   // (1024,) f32
    const int*   bidx = (const int*)d_in[3];     // (8,512) int
    const int*   pxid = (const int*)d_in[4];     // (8,512) int
    const int*   pyid = (const int*)d_in[5];     // (8,512) int
    float* out = (float*)d_out;                  // (8,4096,512) f32

    dim3 grid(1024, 4, 1);   // 1024 sample tiles x 4 filter blocks
    dim3 block(256, 1, 1);   // 8 waves
    hipLaunchKernelGGL(index2feat_wmma, grid, block, 0, stream,
                       img, Wm, bias, bidx, pxid, pyid, out);
}